// DIFT_linear_projection_46617575031122
// MI455X (gfx1250) — compile-verified
//
#include <hip/hip_runtime.h>
#include <math.h>

typedef __attribute__((ext_vector_type(2))) float v2f;
typedef __attribute__((ext_vector_type(8))) float v8f;

// explicit global-address-space pointer so the K-loop lowers to
// global_load_b32 (LOADcnt only) instead of flat_load_b32 (LOADcnt+DScnt)
typedef const float __attribute__((address_space(1)))* gptr;
#define AS_GLOBAL(p) ((gptr)(uintptr_t)(p))

#define B_SZ    256
#define M_SZ    64
#define L_SZ    24576
#define KSPLIT  32                    // total K partitions
#define KWAVES  8                     // waves per block (in-block K split)
#define KBLOCKS (KSPLIT / KWAVES)     // 4 grid.z K-slices (atomic-reduced)
#define KW      (L_SZ / KSPLIT)       // 768 K elements per wave

// ---------------------------------------------------------------------------
// Kernel 1: inv-norm per (m,i) column of the kernel tensor, plus zero d_out.
// grid = 192 blocks (one per m*3+i), block = 256 threads.
// 192*256 == 49152 == out_size, so this kernel also zero-fills d_out for the
// atomic-accumulate epilogue of kernel 2.
// ---------------------------------------------------------------------------
__global__ void __launch_bounds__(256)
norm_zero_kernel(const float* __restrict__ kern,   // [M, L, 3]
                 float* __restrict__ invnorm,      // [192] in d_ws
                 float* __restrict__ out)          // [B, M, 3]
{
    const int mi = blockIdx.x;          // m*3 + i
    const int m  = mi / 3;
    const int i  = mi % 3;
    gptr p = AS_GLOBAL(kern + (size_t)m * (3u * L_SZ) + i);

    float s = 0.0f;
    for (int l = threadIdx.x; l < L_SZ; l += 256) {
        float v = p[3 * l];
        s += v * v;
    }

    __shared__ float red[256];
    red[threadIdx.x] = s;
    __syncthreads();
    #pragma unroll
    for (int off = 128; off > 0; off >>= 1) {
        if (threadIdx.x < off) red[threadIdx.x] += red[threadIdx.x + off];
        __syncthreads();
    }
    if (threadIdx.x == 0) {
        float n = sqrtf(red[0]);
        invnorm[mi] = 1.0f / fmaxf(n, 1e-12f);
    }

    out[blockIdx.x * 256 + threadIdx.x] = 0.0f;
}

// ---------------------------------------------------------------------------
// Kernel 2: G[192,768] = A'[192,K] x B'[K,768] via V_WMMA_F32_16X16X4_F32.
//   A'[m*3+i, l] = kern[m,l,i]   (K-stride 3 floats)
//   B'[l, b*3+c] = lumi[b,l,c]   (K-stride 3 floats)
// Each wave computes a full 32x64 region (2x4 tiles of 16x16) -> 12 loads
// per 8 WMMAs (1.5 loads/WMMA).  8 waves per block split K (768 each) and
// reduce through LDS; grid.z = 4 more K-slices reduce through the atomic
// epilogue (all scaling factors are linear, so partials distribute).
// grid = (12 n-regions of 64, 6 m-regions of 32, 4 K-slices), block = 256.
// ---------------------------------------------------------------------------
__global__ void __launch_bounds__(256)
gemm_wmma_kernel(const float* __restrict__ lumi,    // [B, L, 3]
                 const float* __restrict__ kern,    // [M, L, 3]
                 const float* __restrict__ rgb,     // [9, 3]
                 const float* __restrict__ noise,   // [B, M, 3]
                 const float* __restrict__ invnorm, // [192]
                 float* __restrict__ out)           // [B, M, 3]
{
    const int n_reg = blockIdx.x;    // 0..11 (64 cols each)
    const int m_reg = blockIdx.y;    // 0..5  (32 rows each)
    const int k_reg = blockIdx.z;    // 0..3
    const int tid  = threadIdx.x;
    const int wave = tid >> 5;
    const int lane = tid & 31;

    // ISA 16x4 f32 A fragment: lanes 0-15 -> M=lane, K={0,1}; lanes 16-31 ->
    // M=lane-16, K={2,3}. B fragment mirrored on N.
    const int frow = lane & 15;
    const int kb   = (lane >> 4) << 1;         // 0 or 2

    const size_t k0 = (size_t)((k_reg * KWAVES + wave) * KW + kb);

    // 2 A rows (tr=0,1), 4 B cols (tc=0..3) per lane — scalar globals
    const int mrow0 = m_reg * 32 + frow;             // m' = 3m + i
    const int mrow1 = mrow0 + 16;
    const int ncol0 = n_reg * 64 + frow;             // n' = 3b + c
    const int ncol1 = ncol0 + 16;
    const int ncol2 = ncol0 + 32;
    const int ncol3 = ncol0 + 48;

    gptr ap0 = AS_GLOBAL(kern + (size_t)(mrow0 / 3) * (3u * L_SZ) + (mrow0 % 3) + 3 * k0);
    gptr ap1 = AS_GLOBAL(kern + (size_t)(mrow1 / 3) * (3u * L_SZ) + (mrow1 % 3) + 3 * k0);
    gptr bp0 = AS_GLOBAL(lumi + (size_t)(ncol0 / 3) * (3u * L_SZ) + (ncol0 % 3) + 3 * k0);
    gptr bp1 = AS_GLOBAL(lumi + (size_t)(ncol1 / 3) * (3u * L_SZ) + (ncol1 % 3) + 3 * k0);
    gptr bp2 = AS_GLOBAL(lumi + (size_t)(ncol2 / 3) * (3u * L_SZ) + (ncol2 % 3) + 3 * k0);
    gptr bp3 = AS_GLOBAL(lumi + (size_t)(ncol3 / 3) * (3u * L_SZ) + (ncol3 % 3) + 3 * k0);

    v8f acc00 = {}, acc01 = {}, acc02 = {}, acc03 = {};
    v8f acc10 = {}, acc11 = {}, acc12 = {}, acc13 = {};

    // 96 outer iterations; 2 K-steps (16 WMMAs, 24 loads) per iteration with
    // compile-time element offsets -> immediate-offset global_load_b32s.
    for (int kk = 0; kk < KW; kk += 8) {
        #pragma unroll
        for (int u = 0; u < 2; ++u) {
            v2f a0, a1, b0, b1, b2, b3;
            a0.x = ap0[12 * u];  a0.y = ap0[12 * u + 3];
            a1.x = ap1[12 * u];  a1.y = ap1[12 * u + 3];
            b0.x = bp0[12 * u];  b0.y = bp0[12 * u + 3];
            b1.x = bp1[12 * u];  b1.y = bp1[12 * u + 3];
            b2.x = bp2[12 * u];  b2.y = bp2[12 * u + 3];
            b3.x = bp3[12 * u];  b3.y = bp3[12 * u + 3];
            acc00 = __builtin_amdgcn_wmma_f32_16x16x4_f32(false, a0, false, b0, (short)0, acc00, false, false);
            acc01 = __builtin_amdgcn_wmma_f32_16x16x4_f32(false, a0, false, b1, (short)0, acc01, false, false);
            acc02 = __builtin_amdgcn_wmma_f32_16x16x4_f32(false, a0, false, b2, (short)0, acc02, false, false);
            acc03 = __builtin_amdgcn_wmma_f32_16x16x4_f32(false, a0, false, b3, (short)0, acc03, false, false);
            acc10 = __builtin_amdgcn_wmma_f32_16x16x4_f32(false, a1, false, b0, (short)0, acc10, false, false);
            acc11 = __builtin_amdgcn_wmma_f32_16x16x4_f32(false, a1, false, b1, (short)0, acc11, false, false);
            acc12 = __builtin_amdgcn_wmma_f32_16x16x4_f32(false, a1, false, b2, (short)0, acc12, false, false);
            acc13 = __builtin_amdgcn_wmma_f32_16x16x4_f32(false, a1, false, b3, (short)0, acc13, false, false);
        }
        ap0 += 24; ap1 += 24;
        bp0 += 24; bp1 += 24; bp2 += 24; bp3 += 24;
    }

    // cross-wave K reduction through LDS: 8 waves x 8 tiles x 1KB = 64 KB
    __shared__ float red[KWAVES * 8 * 256];
    {
        float* wr = red + wave * 2048;
        #pragma unroll
        for (int j = 0; j < 8; ++j) {
            wr[0 * 256 + j * 32 + lane] = acc00[j];
            wr[1 * 256 + j * 32 + lane] = acc01[j];
            wr[2 * 256 + j * 32 + lane] = acc02[j];
            wr[3 * 256 + j * 32 + lane] = acc03[j];
            wr[4 * 256 + j * 32 + lane] = acc10[j];
            wr[5 * 256 + j * 32 + lane] = acc11[j];
            wr[6 * 256 + j * 32 + lane] = acc12[j];
            wr[7 * 256 + j * 32 + lane] = acc13[j];
        }
    }
    __syncthreads();

    // 2048 G elements per block, 256 threads -> 8 elements per thread.
    #pragma unroll
    for (int s = 0; s < 8; ++s) {
        const int e   = tid + s * 256;        // 0..2047
        const int tt  = e >> 8;               // tile: tr = tt>>2, tc = tt&3
        const int idx = e & 255;              // element within 16x16 tile

        float g = 0.0f;
        #pragma unroll
        for (int w = 0; w < KWAVES; ++w)
            g += red[w * 2048 + e];

        // C/D layout: idx = j*32 + t ; M = j + 8*(t>>4), N = t&15
        const int j   = idx >> 5;
        const int t   = idx & 31;
        const int row = j + 8 * (t >> 4);
        const int col = t & 15;
        const int gm  = m_reg * 32 + (tt >> 2) * 16 + row;   // 3m + i
        const int gn  = n_reg * 64 + (tt & 3) * 16 + col;    // 3b + c
        const int om  = gm / 3, oi = gm % 3;
        const int ob  = gn / 3, oc = gn % 3;

        const float val = g * invnorm[gm];
        const float* nz = noise + (size_t)ob * (M_SZ * 3) + om * 3;
        float*       op = out   + (size_t)ob * (M_SZ * 3) + om * 3;

        #pragma unroll
        for (int cp = 0; cp < 3; ++cp) {
            // noise factor multiplies the final sum -> distributes over terms
            float w = rgb[(oi * 3 + oc) * 3 + cp] * (1.0f + 0.01f * nz[cp]);
            atomicAdd(&op[cp], val * w);
        }
    }
}

// ---------------------------------------------------------------------------
extern "C" void kernel_launch(void* const* d_in, const int* in_sizes, int n_in,
                              void* d_out, int out_size, void* d_ws, size_t ws_size,
                              hipStream_t stream)
{
    const float* lumi  = (const float*)d_in[0];   // [256, 24576, 3]
    const float* kern  = (const float*)d_in[1];   // [64, 24576, 3]
    const float* rgb   = (const float*)d_in[2];   // [9, 3]
    const float* noise = (const float*)d_in[3];   // [256, 64, 3]
    float* out = (float*)d_out;                   // [256, 64, 3]
    float* invnorm = (float*)d_ws;                // 192 floats of scratch

    // pass 1: column norms of kernel + zero the output
    norm_zero_kernel<<<192, 256, 0, stream>>>(kern, invnorm, out);

    // pass 2: WMMA GEMM (2x4 register tiles, 32-way K split) + fused epilogue
    dim3 grid(12, 6, KBLOCKS);
    gemm_wmma_kernel<<<grid, 256, 0, stream>>>(lumi, kern, rgb, noise,
                                               invnorm, out);
}